// DTWLoss_32280974197174
// MI455X (gfx1250) — compile-verified
//
#include <hip/hip_runtime.h>
#include <hip/hip_bf16.h>

typedef __attribute__((ext_vector_type(2))) float v2f;
typedef __attribute__((ext_vector_type(8))) float v8f;

#define Bn 64
#define S  512
#define F  128
#define BIGV 1e30f

// ---------------------------------------------------------------------------
// Kernel 1: squared row norms.  One wave32 per row, 4 elems/lane, shfl reduce.
// grid = (Bn*S/8, 2) ; block = 256 (8 waves). y==0 -> pred/xnorm, y==1 -> target/ynorm
// ---------------------------------------------------------------------------
__global__ __launch_bounds__(256) void dtw_row_norms(const float* __restrict__ pred,
                                                     const float* __restrict__ target,
                                                     float* __restrict__ xnorm,
                                                     float* __restrict__ ynorm) {
    const int lane = threadIdx.x & 31;
    const int wave = threadIdx.x >> 5;
    const int row  = blockIdx.x * 8 + wave;              // 0 .. Bn*S-1
    const float* src = (blockIdx.y == 0) ? pred : target;
    float*       dst = (blockIdx.y == 0) ? xnorm : ynorm;
    const float* p = src + (size_t)row * F;
    float s = 0.f;
#pragma unroll
    for (int k = 0; k < F; k += 32) { float v = p[k + lane]; s += v * v; }
#pragma unroll
    for (int off = 16; off; off >>= 1) s += __shfl_down(s, off, 32);
    if (lane == 0) dst[row] = s;
}

// ---------------------------------------------------------------------------
// Kernel 2: cost[b][i][j] = sqrt(max(|x_i|^2 + |y_j|^2 - 2*dot(x_i,y_j), 0))
// Exact-f32 GEMM via V_WMMA_F32_16X16X4_F32.  One wave per 16x16 output tile,
// K=128 -> 32 chained WMMAs.  8 waves / block, 65536 tiles total -> 8192 blocks.
//
// f32 16x16x4 fragment layout (ISA 7.12.2): A lane l holds M=l&15, K = 2*(l>>4)
// and K+1 (one float2).  B lane l holds N=l&15, K = 2*(l>>4) and K+1 -> identical
// float2 addressing on the row-major target.  D: acc[v] = (M = v + 8*(l>=16), N=l&15).
// ---------------------------------------------------------------------------
__global__ __launch_bounds__(256) void dtw_cost_gemm(const float* __restrict__ pred,
                                                     const float* __restrict__ target,
                                                     const float* __restrict__ xnorm,
                                                     const float* __restrict__ ynorm,
                                                     float* __restrict__ cost) {
    const int lane = threadIdx.x & 31;
    const int wave = threadIdx.x >> 5;
    const int t    = blockIdx.x * 8 + wave;   // global tile id
    const int b    = t >> 10;                 // 1024 tiles per batch (32x32)
    const int tile = t & 1023;
    const int i0   = (tile >> 5) << 4;
    const int j0   = (tile & 31) << 4;

    const float* X = pred   + (size_t)b * S * F;
    const float* Y = target + (size_t)b * S * F;

    const int r  = lane & 15;            // row within fragment
    const int ko = (lane >> 4) << 1;     // K sub-offset: 0 or 2

    const float* xrow = X + (size_t)(i0 + r) * F + ko;
    const float* yrow = Y + (size_t)(j0 + r) * F + ko;

    v8f acc = {};
#pragma unroll
    for (int k = 0; k < F; k += 4) {
        v2f a  = *(const v2f*)(xrow + k);
        v2f bb = *(const v2f*)(yrow + k);
        acc = __builtin_amdgcn_wmma_f32_16x16x4_f32(false, a, false, bb,
                                                    (short)0, acc, false, false);
    }

    // epilogue: d2 = |x|^2 + |y|^2 - 2*dot ; cost = sqrt(max(d2,0))
    const int   n     = j0 + r;
    const float yn    = ynorm[b * S + n];
    const int   mbase = i0 + ((lane >> 4) << 3);
    float* crow = cost + (size_t)b * S * S;
#pragma unroll
    for (int v = 0; v < 8; ++v) {
        const int m  = mbase + v;
        float d2 = xnorm[b * S + m] + yn - 2.0f * acc[v];
        crow[(size_t)m * S + n] = __builtin_sqrtf(fmaxf(d2, 0.0f));
    }
}

// ---------------------------------------------------------------------------
// Kernel 3: soft-DTW wavefront DP.  One 512-thread block per batch (16 waves on
// a WGP).  Three rotating diagonal buffers in LDS; barrier per anti-diagonal.
// R is the (S+1)x(S+1) DP table; buf[d%3][i] = R[i][d-i].  BIGV stands in for
// +inf so min/exp stay NaN-free (exp(-1e30) underflows to 0, matching the ref).
// ---------------------------------------------------------------------------
__global__ __launch_bounds__(512) void dtw_softdtw(const float* __restrict__ cost,
                                                   float* __restrict__ vals) {
    __shared__ float buf[3][S + 1];
    const int b   = blockIdx.x;
    const int tid = threadIdx.x;

    for (int idx = tid; idx < 3 * (S + 1); idx += blockDim.x)
        ((float*)buf)[idx] = BIGV;
    __syncthreads();
    if (tid == 0) buf[0][0] = 0.0f;   // R[0][0] on diagonal d=0; d=1 stays BIG
    __syncthreads();

    const float* C = cost + (size_t)b * S * S;
    int cur = 2;
    for (int d = 2; d <= 2 * S; ++d) {
        float*       pc = buf[cur];
        const float* p1 = buf[(cur + 2) % 3];   // diagonal d-1
        const float* p2 = buf[(cur + 1) % 3];   // diagonal d-2
        int lo = d - S; if (lo < 1) lo = 1;
        int hi = d - 1; if (hi > S) hi = S;
        const int i = lo + tid;
        if (i <= hi) {
            const int j = d - i;
            float c    = C[(size_t)(i - 1) * S + (j - 1)];
            float up   = p1[i - 1];   // R[i-1][j]
            float left = p1[i];       // R[i][j-1]
            float dg   = p2[i - 1];   // R[i-1][j-1]
            float m = fminf(up, fminf(left, dg));
            float s = __expf(m - up) + __expf(m - left) + __expf(m - dg);
            pc[i] = c + m - __logf(s);
            // warm L2->L0 path for next diagonal's cost element (same i, j+1)
            if (j < S) __builtin_prefetch(&C[(size_t)(i - 1) * S + j], 0, 0);
        }
        if (d <= S && tid == 0) { pc[0] = BIGV; pc[d] = BIGV; }  // R[0][d], R[d][0]
        cur = cur + 1; if (cur == 3) cur = 0;
        __syncthreads();
    }
    // diagonal d=2S lives in buffer (cur+2)%3 after the final rotate
    if (tid == 0) vals[b] = buf[(cur + 2) % 3][S] * (1.0f / (float)(2 * S));
}

// ---------------------------------------------------------------------------
// Kernel 4: mean over the 64 per-batch values -> scalar output.
// ---------------------------------------------------------------------------
__global__ __launch_bounds__(64) void dtw_reduce(const float* __restrict__ vals,
                                                 float* __restrict__ out) {
    __shared__ float s[Bn];
    const int t = threadIdx.x;
    s[t] = vals[t];
    __syncthreads();
    for (int off = Bn / 2; off > 0; off >>= 1) {
        if (t < off) s[t] += s[t + off];
        __syncthreads();
    }
    if (t == 0) out[0] = s[0] * (1.0f / (float)Bn);
}

// ---------------------------------------------------------------------------
extern "C" void kernel_launch(void* const* d_in, const int* in_sizes, int n_in,
                              void* d_out, int out_size, void* d_ws, size_t ws_size,
                              hipStream_t stream) {
    const float* pred   = (const float*)d_in[0];
    const float* target = (const float*)d_in[1];
    float* out = (float*)d_out;

    // workspace layout: cost (64*512*512) | xnorm (64*512) | ynorm (64*512) | vals (64)
    float* cost  = (float*)d_ws;
    float* xnorm = cost + (size_t)Bn * S * S;
    float* ynorm = xnorm + (size_t)Bn * S;
    float* vals  = ynorm + (size_t)Bn * S;

    dim3 gNorm(Bn * S / 8, 2);
    dtw_row_norms<<<gNorm, 256, 0, stream>>>(pred, target, xnorm, ynorm);

    const int tiles  = Bn * (S / 16) * (S / 16);   // 65536
    dtw_cost_gemm<<<tiles / 8, 256, 0, stream>>>(pred, target, xnorm, ynorm, cost);

    dtw_softdtw<<<Bn, S, 0, stream>>>(cost, vals);

    dtw_reduce<<<1, Bn, 0, stream>>>(vals, out);
}